// VectorQuantizer_45174466020056
// MI455X (gfx1250) — compile-verified
//
#include <hip/hip_runtime.h>

typedef __attribute__((ext_vector_type(2))) float v2f;
typedef __attribute__((ext_vector_type(8))) float v8f;

#define B_  16384
#define D_  512
#define K_  8192
#define BM  128          // z rows per workgroup
#define ZSTR 516         // padded LDS row stride (floats): 4*r mod 64 distinct -> conflict-free
#define NT  (K_ / 16)    // 512 codebook tiles of 16

// workspace layout (float units):
//   [0]                 loss accumulator
//   [16 .. 16+K_)       esq[K_]  (||e_k||^2)
//   [16+K_ .. 16+K_+B_) int indices[B_]
static constexpr size_t LDS_BYTES = (size_t)(BM + 16) * ZSTR * sizeof(float);

// ---------------- Kernel 1: ||e_k||^2 (one warp per codebook row) ------------
__global__ void vq_esq(const float* __restrict__ E, float* __restrict__ esq,
                       float* __restrict__ lossAcc) {
    if (blockIdx.x == 0 && threadIdx.x == 0) *lossAcc = 0.0f;
    const int warp = (blockIdx.x * blockDim.x + threadIdx.x) >> 5;
    const int lane = threadIdx.x & 31;
    const float4* row = (const float4*)(E + (size_t)warp * D_);
    float s = 0.0f;
    for (int i = lane; i < D_ / 4; i += 32) {
        float4 v = row[i];
        s += v.x * v.x + v.y * v.y + v.z * v.z + v.w * v.w;
    }
    #pragma unroll
    for (int off = 16; off; off >>= 1) s += __shfl_xor(s, off, 32);
    if (lane == 0) esq[warp] = s;
}

// ---------------- Kernel 2: fused score GEMM + row argmin (WMMA f32) ---------
__launch_bounds__(256)
__global__ void vq_argmin(const float* __restrict__ Z, const float* __restrict__ E,
                          const float* __restrict__ esq, int* __restrict__ idxOut,
                          float* __restrict__ idxF) {
    extern __shared__ float lds[];
    float* zT = lds;                       // [BM][ZSTR]  persistent z tile
    float* eT = lds + (size_t)BM * ZSTR;   // [16][ZSTR]  streaming E tile

    const int tid     = threadIdx.x;
    const int rowBase = blockIdx.x * BM;

    // Load z tile once (float4 global -> float4 LDS; ZSTR*4 bytes % 16 == 0)
    {
        const float4* zg = (const float4*)(Z + (size_t)rowBase * D_);
        for (int i = tid; i < BM * (D_ / 4); i += 256) {
            int r  = i >> 7;          // D_/4 == 128
            int c4 = i & 127;
            *(float4*)(zT + (size_t)r * ZSTR + c4 * 4) = zg[(size_t)r * (D_ / 4) + c4];
        }
    }
    __syncthreads();

    const int lane = tid & 31;
    const int wave = tid >> 5;
    const int half = lane >> 4;   // 0: K=0,1 pair ; 1: K=2,3 pair (A/B frag layout)
    const int ln   = lane & 15;

    const float* za = zT + (size_t)(wave * 16 + ln) * ZSTR + 2 * half;
    const float* eb = eT + (size_t)ln * ZSTR + 2 * half;

    float minv[8];
    int   mini[8];
    #pragma unroll
    for (int v = 0; v < 8; ++v) { minv[v] = 3.4e38f; mini[v] = 0; }

    for (int kt = 0; kt < NT; ++kt) {
        // Cooperative load of 16 codebook rows into LDS
        const float4* eg = (const float4*)(E + (size_t)kt * 16 * D_);
        for (int i = tid; i < 16 * (D_ / 4); i += 256) {
            int r  = i >> 7;
            int c4 = i & 127;
            *(float4*)(eT + (size_t)r * ZSTR + c4 * 4) = eg[(size_t)r * (D_ / 4) + c4];
        }
        __syncthreads();

        if (kt + 1 < NT) {  // global_prefetch_b8 of next tile (32KB spread over 256 lanes)
            __builtin_prefetch((const char*)(E + (size_t)(kt + 1) * 16 * D_) + tid * 128, 0, 1);
        }

        // 16x16 score tile: accumulate over D in K=4 steps
        v8f c = {0.f, 0.f, 0.f, 0.f, 0.f, 0.f, 0.f, 0.f};
        #pragma unroll 8
        for (int d = 0; d < D_; d += 4) {
            v2f a = *(const v2f*)(za + d);
            v2f b = *(const v2f*)(eb + d);
            c = __builtin_amdgcn_wmma_f32_16x16x4_f32(
                    false, a, false, b, (short)0, c, false, false);
        }

        // dist(b,k) ~ ||e||^2 - 2 z.e   (||z||^2 constant per row -> dropped)
        const int   n  = kt * 16 + ln;
        const float es = esq[n];
        #pragma unroll
        for (int v = 0; v < 8; ++v) {
            float dist = fmaf(-2.0f, c[v], es);
            if (dist < minv[v]) { minv[v] = dist; mini[v] = n; }  // strict < keeps lowest k
        }
        __syncthreads();
    }

    // Reduce (min, argmin) across the 16 lanes holding the same M row
    #pragma unroll
    for (int v = 0; v < 8; ++v) {
        float mv = minv[v];
        int   mi = mini[v];
        #pragma unroll
        for (int off = 1; off <= 8; off <<= 1) {   // stays within each 16-lane half
            float ov = __shfl_xor(mv, off, 32);
            int   oi = __shfl_xor(mi, off, 32);
            if (ov < mv || (ov == mv && oi < mi)) { mv = ov; mi = oi; }
        }
        if (ln == 0) {
            int row = rowBase + wave * 16 + half * 8 + v;   // C/D layout: VGPR v -> M=v / v+8
            idxOut[row] = mi;
            idxF[row]   = (float)mi;
        }
    }
}

// ---------------- Kernel 3: gather + straight-through + commitment loss ------
__global__ void vq_gather(const float* __restrict__ Z, const float* __restrict__ E,
                          const int* __restrict__ idx, float* __restrict__ outQ,
                          float* __restrict__ lossAcc) {
    const size_t g  = (size_t)blockIdx.x * 256 + threadIdx.x;  // float4 index
    const int    b  = (int)(g >> 7);                           // 128 float4 per row
    const int    c4 = (int)(g & 127);
    const int    k  = idx[b];

    const float4 q = ((const float4*)(E + (size_t)k * D_))[c4];
    const float4 z = ((const float4*)Z)[g];

    float4 o;   // z + stop_grad(q - z): reproduce the reference expression exactly
    o.x = z.x + (q.x - z.x);
    o.y = z.y + (q.y - z.y);
    o.z = z.z + (q.z - z.z);
    o.w = z.w + (q.w - z.w);
    ((float4*)outQ)[g] = o;

    float dx = z.x - q.x, dy = z.y - q.y, dz = z.z - q.z, dw = z.w - q.w;
    float s = dx * dx + dy * dy + dz * dz + dw * dw;
    #pragma unroll
    for (int off = 16; off; off >>= 1) s += __shfl_xor(s, off, 32);
    if ((threadIdx.x & 31) == 0) atomicAdd(lossAcc, s);
}

// ---------------- Kernel 4: finalize loss ------------------------------------
__global__ void vq_finalize(const float* __restrict__ lossAcc, float* __restrict__ outLoss) {
    *outLoss = 0.25f * (*lossAcc) / (float)((size_t)B_ * D_);
}

extern "C" void kernel_launch(void* const* d_in, const int* in_sizes, int n_in,
                              void* d_out, int out_size, void* d_ws, size_t ws_size,
                              hipStream_t stream) {
    const float* Z = (const float*)d_in[0];   // [B_, D_] f32
    const float* E = (const float*)d_in[1];   // [K_, D_] f32

    float* out      = (float*)d_out;
    float* outQ     = out;                    // [B_*D_] quantized_st
    float* outIdxF  = out + (size_t)B_ * D_;  // [B_]    indices (as float)
    float* outLoss  = outIdxF + B_;           // [1]     loss

    float* ws       = (float*)d_ws;
    float* lossAcc  = ws;                     // 1 float
    float* esq      = ws + 16;                // K_ floats
    int*   idxWs    = (int*)(ws + 16 + K_);   // B_ ints

    // 1) ||e_k||^2 (+ zero loss accumulator)
    vq_esq<<<(K_ * 32) / 256, 256, 0, stream>>>(E, esq, lossAcc);

    // 2) fused f32-WMMA score GEMM + argmin (128 blocks, 8 waves, 290KB LDS/WGP)
    vq_argmin<<<B_ / BM, 256, LDS_BYTES, stream>>>(Z, E, esq, idxWs, outIdxF);

    // 3) gather + straight-through output + commitment-loss partial sums
    vq_gather<<<((size_t)B_ * D_ / 4) / 256, 256, 0, stream>>>(Z, E, idxWs, outQ, lossAcc);

    // 4) finalize scalar loss
    vq_finalize<<<1, 1, 0, stream>>>(lossAcc, outLoss);
}